// CAREConv_21045339750810
// MI455X (gfx1250) — compile-verified
//
#include <hip/hip_runtime.h>
#include <math.h>

typedef __attribute__((ext_vector_type(2))) float v2f;
typedef __attribute__((ext_vector_type(8))) float v8f;

#define DEG    32
#define KSEL   16
#define IN_F   128
#define H_F    64
#define P_KEEP 0.5f
#define HPAD   132   // LDS row stride (floats) to break bank conflicts

// ---------------------------------------------------------------------------
// Kernel 1: t[n, 0:2] = tanh(feature[n,:] @ W_mlp^T + b_mlp), one wave per node
// ---------------------------------------------------------------------------
__global__ __launch_bounds__(256) void care_mlp_tanh(
    const float* __restrict__ feature, const float* __restrict__ W_mlp,
    const float* __restrict__ b_mlp, float* __restrict__ t_out, int n_nodes) {
  const int wave = (int)((blockIdx.x * blockDim.x + threadIdx.x) >> 5);
  const int lane = (int)(threadIdx.x & 31);
  if (wave >= n_nodes) return;

  const float4 f  = *(const float4*)(feature + wave * IN_F + lane * 4);
  const float4 w0 = *(const float4*)(W_mlp + lane * 4);
  const float4 w1 = *(const float4*)(W_mlp + IN_F + lane * 4);
  float p0 = f.x * w0.x + f.y * w0.y + f.z * w0.z + f.w * w0.w;
  float p1 = f.x * w1.x + f.y * w1.y + f.z * w1.z + f.w * w1.w;
#pragma unroll
  for (int off = 16; off; off >>= 1) {
    p0 += __shfl_xor(p0, off, 32);
    p1 += __shfl_xor(p1, off, 32);
  }
  if (lane == 0) {
    float2 t;
    t.x = tanhf(p0 + b_mlp[0]);
    t.y = tanhf(p1 + b_mlp[1]);
    *(float2*)(t_out + wave * 2) = t;
  }
}

// ---------------------------------------------------------------------------
// Kernel 2: distance -> exact top-16 (rank count, top_k tie-break) -> mean
// gather -> residual -> h rows to LDS -> fp32 WMMA GEMM h @ W_lin^T + b_lin.
// 512 threads = 16 waves = 16 nodes per block; waves 0..3 do 4 N-tiles.
// ---------------------------------------------------------------------------
__global__ __launch_bounds__(512) void care_agg_gemm(
    const float* __restrict__ feature, const int* __restrict__ src_ids,
    const float* __restrict__ t, const float* __restrict__ W_lin,
    const float* __restrict__ b_lin, float* __restrict__ out, int n_nodes) {
  __shared__ float hLDS[16 * HPAD];

  const int wave = (int)(threadIdx.x >> 5);
  const int lane = (int)(threadIdx.x & 31);
  const int base = (int)blockIdx.x * 16;
  const int node = base + wave;

  float4 h = {0.f, 0.f, 0.f, 0.f};
  if (node < n_nodes) {
    // --- per-edge L1 distance on 2-dim tanh-MLP outputs (lane = edge) ---
    const int    src = src_ids[node * DEG + lane];
    const float2 ts  = *(const float2*)(t + src * 2);
    const float2 td  = *(const float2*)(t + node * 2);
    const float  d   = fabsf(ts.x - td.x) + fabsf(ts.y - td.y);

    // --- exact top-K=16 smallest: rank by (d, lane) lexicographic order ---
    int rank = 0;
#pragma unroll
    for (int j = 0; j < DEG; ++j) {
      const float dj = __shfl(d, j, 32);
      rank += (dj < d || (dj == d && j < lane)) ? 1 : 0;
    }
    unsigned mask = __builtin_amdgcn_ballot_w32(rank < KSEL);  // uniform, 16 bits set

    // --- mean-aggregate the 16 selected rows (coalesced 512B gathers; L2-hot)
    float4 acc = {0.f, 0.f, 0.f, 0.f};
    while (mask) {
      const int e = __ffs(mask) - 1;
      mask &= mask - 1;
      const int s = __shfl(src, e, 32);
      const float4 f = *(const float4*)(feature + s * IN_F + lane * 4);
      acc.x += f.x; acc.y += f.y; acc.z += f.z; acc.w += f.w;
    }
    const float4 fi = *(const float4*)(feature + node * IN_F + lane * 4);
    const float  sc = P_KEEP / (float)KSEL;  // P * mean
    h.x = fi.x + sc * acc.x;
    h.y = fi.y + sc * acc.y;
    h.z = fi.z + sc * acc.z;
    h.w = fi.w + sc * acc.w;
  }
  // row `wave`, cols [4*lane, 4*lane+3]; stride HPAD=132 keeps column reads
  // in phase 2 on distinct LDS banks.
  *(float4*)(&hLDS[wave * HPAD + lane * 4]) = h;
  __syncthreads();

  // --- phase 2: D[16x64] = A[16x128] * B[128x64] via V_WMMA_F32_16X16X4_F32 ---
  if (wave < 4) {
    const int nTile = wave;
    const int nl    = lane & 15;
    const int hi    = lane >> 4;            // half-wave: K sub-offset 0 or 2
    const int col   = nTile * 16 + nl;      // output column (0..63)

    v8f c = {};
#pragma unroll 4
    for (int kt = 0; kt < IN_F / 4; ++kt) {
      const int k = kt * 4 + hi * 2;
      // A fragment: lanes 0-15 hold (M=nl, K=k..k+1), lanes 16-31 (M=nl, K=k+2..k+3)
      const float2 av = *(const float2*)&hLDS[nl * HPAD + k];
      // B fragment: B[K][N] = W_lin[N][K] (row-major [64,128])
      const float2 bv = *(const float2*)(W_lin + col * IN_F + k);
      v2f a = {av.x, av.y};
      v2f b = {bv.x, bv.y};
      c = __builtin_amdgcn_wmma_f32_16x16x4_f32(
          /*neg_a=*/false, a, /*neg_b=*/false, b,
          /*c_mod=*/(short)0, c, /*reuse_a=*/false, /*reuse_b=*/false);
    }

    // C/D layout: VGPR r -> M = r (lanes 0-15) / r+8 (lanes 16-31), N = lane%16
    const float bl   = b_lin[col];
    const int   mro  = hi * 8;              // row offset for this half-wave
    // Uniform full-tile test (always true when N % 16 == 0): lets the
    // compiler emit 8 unguarded, clause-able global_store_b32 instead of
    // 8 exec-mask toggles + branches.
    if (base + 16 <= n_nodes) {
#pragma unroll
      for (int r = 0; r < 8; ++r) {
        out[(base + mro + r) * H_F + col] = c[r] + bl;
      }
    } else {
#pragma unroll
      for (int r = 0; r < 8; ++r) {
        const int onode = base + mro + r;
        if (onode < n_nodes) out[onode * H_F + col] = c[r] + bl;
      }
    }
  }
}

// ---------------------------------------------------------------------------
extern "C" void kernel_launch(void* const* d_in, const int* in_sizes, int n_in,
                              void* d_out, int out_size, void* d_ws, size_t ws_size,
                              hipStream_t stream) {
  const float* feature = (const float*)d_in[0];
  const int*   src_ids = (const int*)d_in[1];
  const float* W_mlp   = (const float*)d_in[2];
  const float* b_mlp   = (const float*)d_in[3];
  const float* W_lin   = (const float*)d_in[4];
  const float* b_lin   = (const float*)d_in[5];
  float*       out     = (float*)d_out;
  float*       t_ws    = (float*)d_ws;            // N*2 floats (800 KB)

  const int n = in_sizes[0] / IN_F;               // N nodes

  // Kernel 1: one wave per node, 8 waves (256 threads) per block
  const int g1 = (n + 7) / 8;
  care_mlp_tanh<<<g1, 256, 0, stream>>>(feature, W_mlp, b_mlp, t_ws, n);

  // Kernel 2: 16 nodes per block (16 waves)
  const int g2 = (n + 15) / 16;
  care_agg_gemm<<<g2, 512, 0, stream>>>(feature, src_ids, t_ws, W_lin, b_lin,
                                        out, n);
}